// PermutoEncoding_21242908246581
// MI455X (gfx1250) — compile-verified
//
#include <hip/hip_runtime.h>
#include <stdint.h>

typedef __attribute__((ext_vector_type(2))) float    v2f;
typedef __attribute__((ext_vector_type(4))) uint32_t v4u;
typedef __attribute__((ext_vector_type(4))) int32_t  v4i;
typedef __attribute__((ext_vector_type(8))) int32_t  v8i;

#define PERM_CAP  (1u << 19)
#define PERM_MASK (PERM_CAP - 1u)
#define HMULT     2531011u

// One thread = one (point, level) task. 256-thread block = 16 points x 16 levels.
__global__ __launch_bounds__(256) void permuto_fwd_kernel(
    const float* __restrict__ pos,
    const float* __restrict__ params,
    float* __restrict__ out,
    int n_points)
{
    __shared__ float s_pos[16 * 3];
    __shared__ float s_scale[16];

    const int tid    = threadIdx.x;
    const int n_base = blockIdx.x * 16;
    int npts = n_points - n_base;
    if (npts > 16) npts = 16;

    // ---- Stage this block's 16 positions (48 floats) into LDS via the TDM ----
    if (tid == 0) {
        const uint64_t gaddr = (uint64_t)(uintptr_t)(pos + (size_t)n_base * 3);
        const uint32_t laddr = (uint32_t)(uintptr_t)(&s_pos[0]); // low 32 bits = LDS byte offset
        const uint32_t ne    = (uint32_t)(npts * 3);             // elements (4B each)
        v4u g0;
        g0.x = 1u;                                               // count=1, gather off
        g0.y = laddr;                                            // lds_addr
        g0.z = (uint32_t)gaddr;                                  // global_addr[31:0]
        g0.w = ((uint32_t)(gaddr >> 32) & 0x01FFFFFFu)           // global_addr[56:32]
               | (2u << 30);                                     // type = 2 ("image")
        v8i g1;
        g1[0] = (int)(2u << 16);       // workgroup_mask=0, data_size=2 (4 bytes)
        g1[1] = (int)(ne << 16);       // tensor_dim0[15:0]
        g1[2] = (int)(1u << 16);       // tensor_dim0[31:16]=0, tensor_dim1=1
        g1[3] = (int)(ne << 16);       // tile_dim0
        g1[4] = 1;                     // tile_dim1=1, tile_dim2=0
        g1[5] = (int)ne;               // tensor_dim0_stride[31:0]
        g1[6] = (int)(ne << 16);       // tensor_dim1_stride[15:0]
        g1[7] = 0;
        v4i gz4 = {0, 0, 0, 0};
        v8i gz8 = {0, 0, 0, 0, 0, 0, 0, 0};
        __builtin_amdgcn_tensor_load_to_lds(g0, g1, gz4, gz4, gz8, 0);
    }

    // ---- While the TDM copy is in flight: lanes 0-15 of wave 0 build the
    //      scale table (fp64 exp2 executed 16x per block instead of 256x).
    //      scales[l] = 16 * 128^(l/15), double then cast, matching the reference.
    if (tid < 16) {
        s_scale[tid] = (float)(16.0 * exp2(7.0 * (double)tid / 15.0));
    }

    __builtin_amdgcn_s_wait_tensorcnt(0);
    __syncthreads();

    const int ln = tid >> 4;   // local point 0..15
    const int l  = tid & 15;   // level 0..15
    const int n  = n_base + ln;
    if (n >= n_points) return;

    const float sc = s_scale[l];
    const float x = s_pos[ln * 3 + 0] * sc;
    const float y = s_pos[ln * 3 + 1] * sc;
    const float z = s_pos[ln * 3 + 2] * sc;

    // Elevation E (4x3), sf = 1/sqrt(i*(i+1))
    const float sf0 = 0.707106781186547524f;  // 1/sqrt(2)
    const float sf1 = 0.408248290463863016f;  // 1/sqrt(6)
    const float sf2 = 0.288675134594812882f;  // 1/sqrt(12)
    const float e0 = fmaf( sf0, x, fmaf(sf1, y, sf2 * z));
    const float e1 = fmaf(-sf0, x, fmaf(sf1, y, sf2 * z));
    const float e2 = fmaf(-2.0f * sf1, y, sf2 * z);
    const float e3 = -3.0f * sf2 * z;

    // Nearest multiple of 4 (round-half-even, matches jnp.round)
    const float r0 = rintf(e0 * 0.25f) * 4.0f;
    const float r1 = rintf(e1 * 0.25f) * 4.0f;
    const float r2 = rintf(e2 * 0.25f) * 4.0f;
    const float r3 = rintf(e3 * 0.25f) * 4.0f;
    const int sums = (int)((r0 + r1 + r2 + r3) * 0.25f);

    const float d0 = e0 - r0, d1 = e1 - r1, d2 = e2 - r2, d3 = e3 - r3;
    // rank[i] = #{j: d[j]>d[i] or (d[j]==d[i] and j<i)} + sums
    int rank0 = (d1 >  d0) + (d2 >  d0) + (d3 >  d0) + sums;
    int rank1 = (d0 >= d1) + (d2 >  d1) + (d3 >  d1) + sums;
    int rank2 = (d0 >= d2) + (d1 >= d2) + (d3 >  d2) + sums;
    int rank3 = (d0 >= d3) + (d1 >= d3) + (d2 >= d3) + sums;

    int m0 = (int)r0, m1 = (int)r1, m2 = (int)r2, m3 = (int)r3;
    int c;
    c = (rank0 < 0) - (rank0 > 3); rank0 += 4 * c; m0 += 4 * c;
    c = (rank1 < 0) - (rank1 > 3); rank1 += 4 * c; m1 += 4 * c;
    c = (rank2 < 0) - (rank2 > 3); rank2 += 4 * c; m2 += 4 * c;
    c = (rank3 < 0) - (rank3 > 3); rank3 += 4 * c; m3 += 4 * c;

    const float dl0 = (e0 - (float)m0) * 0.25f;
    const float dl1 = (e1 - (float)m1) * 0.25f;
    const float dl2 = (e2 - (float)m2) * 0.25f;
    const float dl3 = (e3 - (float)m3) * 0.25f;

    // Scatter delta by rank (rank is a permutation of {0..3} after clamping)
    const float b0 = (rank0 == 0) ? dl0 : (rank1 == 0) ? dl1 : (rank2 == 0) ? dl2 : dl3;
    const float b1 = (rank0 == 1) ? dl0 : (rank1 == 1) ? dl1 : (rank2 == 1) ? dl2 : dl3;
    const float b2 = (rank0 == 2) ? dl0 : (rank1 == 2) ? dl1 : (rank2 == 2) ? dl2 : dl3;
    const float b3 = (rank0 == 3) ? dl0 : (rank1 == 3) ? dl1 : (rank2 == 3) ? dl2 : dl3;

    const float w0 = 1.0f + b3 - b0;
    const float w1 = b2 - b3;
    const float w2 = b1 - b2;
    const float w3 = b0 - b1;

    // 4 hashed corner keys (dims 0..2 only); int32 wrap semantics via uint32
    uint32_t q0, q1, q2;
    const uint32_t h0 = ((((uint32_t)m0) * HMULT + (uint32_t)m1) * HMULT + (uint32_t)m2) * HMULT;
    q0 = (uint32_t)(m0 + 1 - ((rank0 > 2) << 2));
    q1 = (uint32_t)(m1 + 1 - ((rank1 > 2) << 2));
    q2 = (uint32_t)(m2 + 1 - ((rank2 > 2) << 2));
    const uint32_t h1 = ((q0 * HMULT + q1) * HMULT + q2) * HMULT;
    q0 = (uint32_t)(m0 + 2 - ((rank0 > 1) << 2));
    q1 = (uint32_t)(m1 + 2 - ((rank1 > 1) << 2));
    q2 = (uint32_t)(m2 + 2 - ((rank2 > 1) << 2));
    const uint32_t h2 = ((q0 * HMULT + q1) * HMULT + q2) * HMULT;
    q0 = (uint32_t)(m0 + 3 - ((rank0 > 0) << 2));
    q1 = (uint32_t)(m1 + 3 - ((rank1 > 0) << 2));
    q2 = (uint32_t)(m2 + 3 - ((rank2 > 0) << 2));
    const uint32_t h3 = ((q0 * HMULT + q1) * HMULT + q2) * HMULT;

    // 4 x float2 gathers (L2-resident table), then blend
    const v2f* __restrict__ tab = reinterpret_cast<const v2f*>(params) + (size_t)l * PERM_CAP;
    const v2f f0 = tab[h0 & PERM_MASK];
    const v2f f1 = tab[h1 & PERM_MASK];
    const v2f f2 = tab[h2 & PERM_MASK];
    const v2f f3 = tab[h3 & PERM_MASK];

    float ox = w0 * f0.x, oy = w0 * f0.y;
    ox = fmaf(w1, f1.x, ox); oy = fmaf(w1, f1.y, oy);
    ox = fmaf(w2, f2.x, ox); oy = fmaf(w2, f2.y, oy);
    ox = fmaf(w3, f3.x, ox); oy = fmaf(w3, f3.y, oy);

    v2f r; r.x = ox; r.y = oy;
    // Streaming store: keep L2 capacity for the hash table
    __builtin_nontemporal_store(r, reinterpret_cast<v2f*>(out) + (size_t)n * 16 + l);
}

extern "C" void kernel_launch(void* const* d_in, const int* in_sizes, int n_in,
                              void* d_out, int out_size, void* d_ws, size_t ws_size,
                              hipStream_t stream) {
    const float* pos    = (const float*)d_in[0];
    const float* params = (const float*)d_in[1];
    float* out          = (float*)d_out;
    const int n_points  = in_sizes[0] / 3;
    const int blocks    = (n_points + 15) / 16;  // 16 points x 16 levels per block
    permuto_fwd_kernel<<<blocks, 256, 0, stream>>>(pos, params, out, n_points);
}